// ConvGraphNet_75642964017424
// MI455X (gfx1250) — compile-verified
//
#include <hip/hip_runtime.h>
#include <hip/hip_bf16.h>

typedef float v2f __attribute__((ext_vector_type(2)));
typedef float v8f __attribute__((ext_vector_type(8)));

#define N_NODES 50000
#define N_EDGES 600000
#define FDIM    128

// ---------------------------------------------------------------------------
// Degree / normalization kernels
// ---------------------------------------------------------------------------
__global__ void k_init_deg(float* __restrict__ deg, int n) {
    int i = blockIdx.x * blockDim.x + threadIdx.x;
    if (i < n) deg[i] = 1.0f;                      // self-loop contributes 1
}

__global__ void k_deg_scatter(const int* __restrict__ dst, float* __restrict__ deg, int e) {
    int i = blockIdx.x * blockDim.x + threadIdx.x;
    if (i < e) atomicAdd(&deg[dst[i]], 1.0f);
}

__global__ void k_rsqrt(float* __restrict__ d, int n) {
    int i = blockIdx.x * blockDim.x + threadIdx.x;
    if (i < n) d[i] = __frsqrt_rn(d[i]);           // deg -> deg^{-1/2} in place
}

__global__ void k_edge_norm(const int* __restrict__ src, const int* __restrict__ dst,
                            const float* __restrict__ dis, float* __restrict__ norm, int e) {
    int i = blockIdx.x * blockDim.x + threadIdx.x;
    if (i < e) norm[i] = dis[src[i]] * dis[dst[i]];
}

// ---------------------------------------------------------------------------
// Dense GEMM: Hw[N,128] = Hin[N,128] @ W[128,128] using V_WMMA_F32_16X16X4_F32
// One block = 256 threads = 8 waves; block owns one 16-row stripe, each wave
// owns one 16-col tile. Full W (64 KB) staged in LDS, shared by all 8 waves.
// ---------------------------------------------------------------------------
__global__ __launch_bounds__(256) void k_gemm128_wmma(const float* __restrict__ Hin,
                                                      const float* __restrict__ W,
                                                      float* __restrict__ Hw) {
    __shared__ float sW[FDIM * FDIM];              // 64 KB
    const int t = threadIdx.x;

    // cooperative load of W: 4096 float4, 16 per thread
    const float4* Wv  = (const float4*)W;
    float4*       sWv = (float4*)sW;
#pragma unroll
    for (int i = 0; i < 16; ++i) sWv[t + i * 256] = Wv[t + i * 256];
    __syncthreads();

    const int wave = t >> 5;                       // 0..7 -> column tile
    const int lane = t & 31;
    const int row0 = blockIdx.x * 16;              // 3125 row tiles exactly
    const int col0 = wave * 16;
    const int ln   = lane & 15;                    // A: M index, B: N index
    const int koff = (lane >> 4) << 1;             // 0 (lanes 0-15) / 2 (lanes 16-31)

    const float* arow = Hin + (size_t)(row0 + ln) * FDIM;

    v8f acc = {0.f, 0.f, 0.f, 0.f, 0.f, 0.f, 0.f, 0.f};

#pragma unroll 8
    for (int k0 = 0; k0 < FDIM; k0 += 4) {
        // A 16x4 fp32 layout: lane ln holds (K=koff, K=koff+1) of row M=ln
        v2f a;
        a.x = arow[k0 + koff + 0];
        a.y = arow[k0 + koff + 1];
        // B 4x16 fp32 layout (mirror of A): lane ln holds col N=ln,
        // VGPR pair holds rows K=koff, K=koff+1
        v2f b;
        b.x = sW[(k0 + koff + 0) * FDIM + col0 + ln];
        b.y = sW[(k0 + koff + 1) * FDIM + col0 + ln];
        acc = __builtin_amdgcn_wmma_f32_16x16x4_f32(
            /*neg_a=*/false, a, /*neg_b=*/false, b,
            /*c_mod=*/(short)0, acc, /*reuse_a=*/false, /*reuse_b=*/false);
    }

    // C/D layout: VGPR r -> M = r (lanes 0-15) or M = r+8 (lanes 16-31); N = ln
    const int rbase = (lane >> 4) * 8;
    float* out = Hw + (size_t)(row0 + rbase) * FDIM + col0 + ln;
#pragma unroll
    for (int r = 0; r < 8; ++r) out[(size_t)r * FDIM] = acc[r];
}

// ---------------------------------------------------------------------------
// agg[i,f] = hw[i,f] * dis[i]^2 + bias[f]   (self-loop term + bias as init)
// ---------------------------------------------------------------------------
__global__ void k_agg_init(const float* __restrict__ hw, const float* __restrict__ dis,
                           const float* __restrict__ bias, float* __restrict__ agg, int n) {
    int idx = blockIdx.x * blockDim.x + threadIdx.x;
    if (idx >= n * FDIM) return;
    int i = idx >> 7, f = idx & (FDIM - 1);
    float di = dis[i];
    agg[idx] = hw[idx] * di * di + bias[f];
}

// ---------------------------------------------------------------------------
// Edge scatter: one wave per edge; lane l handles features [4l,4l+4)
// ---------------------------------------------------------------------------
__global__ __launch_bounds__(256) void k_scatter(const float* __restrict__ hw,
                                                 const float* __restrict__ norm,
                                                 const int* __restrict__ src,
                                                 const int* __restrict__ dst,
                                                 float* __restrict__ agg, int e) {
    int wid  = blockIdx.x * (blockDim.x >> 5) + (threadIdx.x >> 5);
    int lane = threadIdx.x & 31;
    if (wid >= e) return;
    int   s  = src[wid];
    int   d  = dst[wid];
    float nm = norm[wid];
    float4 v = ((const float4*)(hw + (size_t)s * FDIM))[lane];
    float* base = agg + (size_t)d * FDIM + lane * 4;
    atomicAdd(base + 0, v.x * nm);
    atomicAdd(base + 1, v.y * nm);
    atomicAdd(base + 2, v.z * nm);
    atomicAdd(base + 3, v.w * nm);
}

__global__ void k_relu(const float* __restrict__ a, float* __restrict__ h, int total) {
    int idx = blockIdx.x * blockDim.x + threadIdx.x;
    if (idx < total) h[idx] = fmaxf(a[idx], 0.0f);
}

// ---------------------------------------------------------------------------
// Layer 3: per-node dot(h[i,:], W3[:,0]) via wave32 shuffle reduction
// ---------------------------------------------------------------------------
__global__ __launch_bounds__(256) void k_dot128(const float* __restrict__ H,
                                                const float* __restrict__ W3,
                                                float* __restrict__ out, int n) {
    int i    = blockIdx.x * (blockDim.x >> 5) + (threadIdx.x >> 5);
    int lane = threadIdx.x & 31;
    if (i >= n) return;
    const float* row = H + (size_t)i * FDIM;
    float s = 0.0f;
#pragma unroll
    for (int c = 0; c < FDIM; c += 32) s = fmaf(row[c + lane], W3[c + lane], s);
#pragma unroll
    for (int off = 16; off > 0; off >>= 1) s += __shfl_down(s, off, 32);
    if (lane == 0) out[i] = s;
}

__global__ void k_out_init(const float* __restrict__ hw3, const float* __restrict__ dis,
                           const float* __restrict__ b3, float* __restrict__ out, int n) {
    int i = blockIdx.x * blockDim.x + threadIdx.x;
    if (i >= n) return;
    float di = dis[i];
    out[i] = hw3[i] * di * di + b3[0];
}

__global__ void k_scatter1(const float* __restrict__ hw3, const float* __restrict__ norm,
                           const int* __restrict__ src, const int* __restrict__ dst,
                           float* __restrict__ out, int e) {
    int i = blockIdx.x * blockDim.x + threadIdx.x;
    if (i < e) atomicAdd(&out[dst[i]], hw3[src[i]] * norm[i]);
}

__global__ void k_relu_inplace(float* __restrict__ out, int n) {
    int i = blockIdx.x * blockDim.x + threadIdx.x;
    if (i < n) out[i] = fmaxf(out[i], 0.0f);
}

// ---------------------------------------------------------------------------
extern "C" void kernel_launch(void* const* d_in, const int* in_sizes, int n_in,
                              void* d_out, int out_size, void* d_ws, size_t ws_size,
                              hipStream_t stream) {
    const float* x   = (const float*)d_in[0];
    const int*   ei  = (const int*)d_in[1];
    const float* W1  = (const float*)d_in[2];
    const float* b1  = (const float*)d_in[3];
    const float* W2  = (const float*)d_in[4];
    const float* b2  = (const float*)d_in[5];
    const float* W3  = (const float*)d_in[6];
    const float* b3  = (const float*)d_in[7];
    float* out = (float*)d_out;

    const int* src = ei;
    const int* dst = ei + N_EDGES;

    // workspace layout (floats); all offsets 16B-aligned
    float* ws   = (float*)d_ws;
    float* dis  = ws;                                   // N
    float* norm = ws + N_NODES;                         // E
    float* hw   = norm + N_EDGES;                       // N*128
    float* agg  = hw + (size_t)N_NODES * FDIM;          // N*128
    float* hbuf = agg + (size_t)N_NODES * FDIM;         // N*128
    float* hw3  = hbuf + (size_t)N_NODES * FDIM;        // N
    (void)ws_size; (void)n_in; (void)in_sizes; (void)out_size;

    const int TB = 256;
    const int gN   = (N_NODES + TB - 1) / TB;           // node-elementwise
    const int gE   = (N_EDGES + TB - 1) / TB;           // edge-elementwise
    const int gNF  = (N_NODES * FDIM + TB - 1) / TB;    // node*feature
    const int gEw  = (N_EDGES + 7) / 8;                 // one wave per edge
    const int gNw  = (N_NODES + 7) / 8;                 // one wave per node
    const int gMM  = N_NODES / 16;                      // 3125 row tiles

    // --- normalization ---
    k_init_deg   <<<gN,  TB, 0, stream>>>(dis, N_NODES);
    k_deg_scatter<<<gE,  TB, 0, stream>>>(dst, dis, N_EDGES);
    k_rsqrt      <<<gN,  TB, 0, stream>>>(dis, N_NODES);
    k_edge_norm  <<<gE,  TB, 0, stream>>>(src, dst, dis, norm, N_EDGES);

    // --- layer 1 ---
    k_gemm128_wmma<<<gMM, TB, 0, stream>>>(x, W1, hw);
    k_agg_init    <<<gNF, TB, 0, stream>>>(hw, dis, b1, agg, N_NODES);
    k_scatter     <<<gEw, TB, 0, stream>>>(hw, norm, src, dst, agg, N_EDGES);
    k_relu        <<<gNF, TB, 0, stream>>>(agg, hbuf, N_NODES * FDIM);

    // --- layer 2 ---
    k_gemm128_wmma<<<gMM, TB, 0, stream>>>(hbuf, W2, hw);
    k_agg_init    <<<gNF, TB, 0, stream>>>(hw, dis, b2, agg, N_NODES);
    k_scatter     <<<gEw, TB, 0, stream>>>(hw, norm, src, dst, agg, N_EDGES);
    k_relu        <<<gNF, TB, 0, stream>>>(agg, hbuf, N_NODES * FDIM);

    // --- layer 3 (D_OUT = 1) ---
    k_dot128      <<<gNw, TB, 0, stream>>>(hbuf, W3, hw3, N_NODES);
    k_out_init    <<<gN,  TB, 0, stream>>>(hw3, dis, b3, out, N_NODES);
    k_scatter1    <<<gE,  TB, 0, stream>>>(hw3, norm, src, dst, out, N_EDGES);
    k_relu_inplace<<<gN,  TB, 0, stream>>>(out, N_NODES);
}